// AFNOBlock_55886114455562
// MI455X (gfx1250) — compile-verified
//
#include <hip/hip_runtime.h>
#include <math.h>

// ---------------- problem constants ----------------
#define HIDDEN 768
#define SEQ    4096
#define BATCH  4
#define NB     8
#define BS     96          // block size = HIDDEN / NB
#define FBINS  2049        // SEQ/2 + 1
#define FPAD   2064        // FBINS padded to multiple of 16
#define MROWS  (BATCH*FPAD)    // 8256 rows for frequency GEMMs
#define TOKENS (BATCH*SEQ)     // 16384 tokens

typedef __attribute__((ext_vector_type(2))) float v2f;
typedef __attribute__((ext_vector_type(8))) float v8f;

// ---------------- WMMA f32 16x16x4 helpers ----------------
// D = A(16x4) * B(4x16) + C, fp32 throughout (exact fp32 semantics).
__device__ __forceinline__ v8f wmma_f32(v2f a, v2f b, v8f c) {
  return __builtin_amdgcn_wmma_f32_16x16x4_f32(
      /*neg_a=*/false, a, /*neg_b=*/false, b,
      /*c_mod=*/(short)0, c, /*reuse_a=*/false, /*reuse_b=*/false);
}

// A fragment: 16x4 tile from row-major [M x lda] source, tile origin at A.
// Lanes 0-15 hold row M=lane, K={0,1}; lanes 16-31 hold row M=lane-16, K={2,3}.
__device__ __forceinline__ v2f load_a_frag(const float* __restrict__ A, int lda) {
  const int lane = threadIdx.x & 31;
  const int row  = lane & 15;
  const int k0   = (lane >> 4) * 2;
  v2f r;
  r.x = A[(size_t)row * lda + k0];
  r.y = A[(size_t)row * lda + k0 + 1];
  return r;
}

// B fragment: 4x16 tile from row-major [K x ldb] source, tile origin at B.
// Lanes 0-15 hold N=lane, K={0,1}; lanes 16-31 hold N=lane-16, K={2,3}.
__device__ __forceinline__ v2f load_b_frag(const float* __restrict__ B, int ldb) {
  const int lane = threadIdx.x & 31;
  const int n  = lane & 15;
  const int k0 = (lane >> 4) * 2;
  v2f r;
  r.x = B[(size_t)k0 * ldb + n];
  r.y = B[(size_t)(k0 + 1) * ldb + n];
  return r;
}

// ---------------- LayerNorm: one wave (32 lanes) per token ----------------
__global__ void layernorm_kernel(const float* __restrict__ X,
                                 const float* __restrict__ G,
                                 const float* __restrict__ Bt,
                                 float* __restrict__ O) {
  const int wave = threadIdx.x >> 5;
  const int lane = threadIdx.x & 31;
  const size_t t = (size_t)blockIdx.x * 8 + wave;
  const float* xp = X + t * HIDDEN;
  float v[24];
  float s = 0.f, s2 = 0.f;
#pragma unroll
  for (int i = 0; i < 24; ++i) {
    float x = xp[lane + i * 32];
    v[i] = x; s += x; s2 += x * x;
  }
#pragma unroll
  for (int o = 16; o > 0; o >>= 1) {
    s  += __shfl_xor(s,  o, 32);
    s2 += __shfl_xor(s2, o, 32);
  }
  const float mu  = s * (1.f / HIDDEN);
  const float var = s2 * (1.f / HIDDEN) - mu * mu;
  const float inv = rsqrtf(var + 1e-5f);
  float* op = O + t * HIDDEN;
#pragma unroll
  for (int i = 0; i < 24; ++i) {
    const int c = lane + i * 32;
    op[c] = (v[i] - mu) * inv * G[c] + Bt[c];
  }
}

// ---------------- radix-2 FFT core in LDS (4096 points, 256 threads) -------
__device__ void fft_core(float* re, float* im, float sgn) {
  const int tid = threadIdx.x;
  for (int len = 2; len <= SEQ; len <<= 1) {
    const int hf = len >> 1;
    const float base = sgn * 6.28318530717958647692f / (float)len;
    for (int j = tid; j < SEQ / 2; j += 256) {
      const int pos = j & (hf - 1);
      const int i1  = ((j & ~(hf - 1)) << 1) | pos;
      const int i2  = i1 + hf;
      float sn, cs;
      __sincosf(base * (float)pos, &sn, &cs);
      const float re2 = re[i2], im2 = im[i2];
      const float tr = cs * re2 - sn * im2;
      const float ti = cs * im2 + sn * re2;
      const float re1 = re[i1], im1 = im[i1];
      re[i2] = re1 - tr; im[i2] = im1 - ti;
      re[i1] = re1 + tr; im[i1] = im1 + ti;
    }
    __syncthreads();
  }
}

// ---------------- forward rFFT (ortho): one workgroup per (b, channel) -----
__global__ void fft_fwd_kernel(const float* __restrict__ XN,
                               float* __restrict__ XR, float* __restrict__ XI) {
  __shared__ float re[SEQ];
  __shared__ float im[SEQ];
  const int h = blockIdx.x;   // channel 0..767
  const int b = blockIdx.y;   // batch
  const int tid = threadIdx.x;
  const float* src = XN + (size_t)b * SEQ * HIDDEN + h;
  for (int i = tid; i < SEQ; i += 256) {
    const int r = __brev((unsigned)i) >> 20;  // 12-bit bit reversal
    re[r] = src[(size_t)i * HIDDEN];
    im[r] = 0.f;
  }
  __syncthreads();
  fft_core(re, im, -1.f);
  const float s = 0.015625f;  // 1/sqrt(4096), ortho
  for (int f = tid; f < FPAD; f += 256) {
    const size_t idx = ((size_t)b * FPAD + f) * HIDDEN + h;
    if (f < FBINS) { XR[idx] = re[f] * s; XI[idx] = im[f] * s; }
    else           { XR[idx] = 0.f;       XI[idx] = 0.f; }
  }
}

// ---------------- inverse rFFT (ortho) + residual --------------------------
__global__ void fft_inv_kernel(const float* __restrict__ R,
                               const float* __restrict__ I,
                               const float* __restrict__ X,
                               float* __restrict__ Yb) {
  __shared__ float re[SEQ];
  __shared__ float im[SEQ];
  const int h = blockIdx.x;
  const int b = blockIdx.y;
  const int tid = threadIdx.x;
  for (int i = tid; i < SEQ; i += 256) {
    const int f = (i <= SEQ / 2) ? i : (SEQ - i);   // Hermitian mirror
    const size_t idx = ((size_t)b * FPAD + f) * HIDDEN + h;
    float vr = R[idx];
    float vi = I[idx];
    if (i > SEQ / 2) vi = -vi;                      // conjugate
    const int r = __brev((unsigned)i) >> 20;
    re[r] = vr; im[r] = vi;
  }
  __syncthreads();
  fft_core(re, im, 1.f);
  const float s = 0.015625f;
  const float* xs = X  + (size_t)b * SEQ * HIDDEN + h;
  float*       ys = Yb + (size_t)b * SEQ * HIDDEN + h;
  for (int i = tid; i < SEQ; i += 256)
    ys[(size_t)i * HIDDEN] = xs[(size_t)i * HIDDEN] + re[i] * s;
}

// ---------------- complex block-diagonal MLP in frequency domain -----------
// O_r = [relu](AR*Wr - AI*Wi + br), O_i = [relu](AI*Wr + AR*Wi + bi)
// One wave per 16x16 output tile; 4 fp32 WMMA accumulators.
__global__ void freq_gemm(const float* __restrict__ AR, const float* __restrict__ AI,
                          const float* __restrict__ Wr, const float* __restrict__ Wi,
                          const float* __restrict__ br, const float* __restrict__ bi,
                          float* __restrict__ OR_, float* __restrict__ OI_,
                          int do_relu) {
  const int wave = threadIdx.x >> 5;
  const int lane = threadIdx.x & 31;
  const int mtile   = blockIdx.x;               // 0..515
  const int colTile = blockIdx.y * 8 + wave;    // 0..47
  const int nb = colTile / 6;                   // which diagonal block
  const int nt = colTile % 6;                   // 16-wide n-tile inside block
  const int m0 = mtile * 16;
  const int c0 = nb * BS;                       // input-column base
  const int n0 = c0 + nt * 16;                  // output-column base

  v8f rr = {}, ri = {}, ir = {}, ii = {};
  const float* arow = AR + (size_t)m0 * HIDDEN + c0;
  const float* airo = AI + (size_t)m0 * HIDDEN + c0;
  const float* wr = Wr + (size_t)nb * BS * BS + nt * 16;
  const float* wi = Wi + (size_t)nb * BS * BS + nt * 16;
#pragma unroll 4
  for (int k = 0; k < BS; k += 4) {
    v2f aR = load_a_frag(arow + k, HIDDEN);
    v2f aI = load_a_frag(airo + k, HIDDEN);
    v2f bR = load_b_frag(wr + (size_t)k * BS, BS);
    v2f bI = load_b_frag(wi + (size_t)k * BS, BS);
    rr = wmma_f32(aR, bR, rr);
    ii = wmma_f32(aI, bI, ii);
    ri = wmma_f32(aR, bI, ri);
    ir = wmma_f32(aI, bR, ir);
  }
  const int n  = lane & 15;
  const int mo = (lane >> 4) * 8;
  const float bre = br[nb * BS + nt * 16 + n];
  const float bim = bi[nb * BS + nt * 16 + n];
#pragma unroll
  for (int v = 0; v < 8; ++v) {
    float orv = rr[v] - ii[v] + bre;
    float oiv = ir[v] + ri[v] + bim;
    if (do_relu) { orv = fmaxf(orv, 0.f); oiv = fmaxf(oiv, 0.f); }
    const size_t idx = (size_t)(m0 + mo + v) * HIDDEN + n0 + n;
    OR_[idx] = orv;
    OI_[idx] = oiv;
  }
}

// ---------------- MLP GEMM 1: O = gelu(A[16384x768] @ W[768x1536] + b) -----
// Wave tile 16x64: one A fragment feeds 4 WMMAs.
__global__ void mlp_gemm1(const float* __restrict__ A,
                          const float* __restrict__ W,
                          const float* __restrict__ bias,
                          float* __restrict__ O) {
  const int wave = threadIdx.x >> 5;
  const int lane = threadIdx.x & 31;
  const int m0 = blockIdx.x * 16;
  const int n0 = (blockIdx.y * 8 + wave) * 64;
  v8f a0 = {}, a1 = {}, a2 = {}, a3 = {};
  const float* arow = A + (size_t)m0 * 768;
  for (int k = 0; k < 768; k += 4) {
    __builtin_prefetch(arow + k + 32, 0, 0);     // global_prefetch_b8
    v2f a = load_a_frag(arow + k, 768);
    const float* wb = W + (size_t)k * 1536 + n0;
    v2f b0 = load_b_frag(wb,      1536);
    v2f b1 = load_b_frag(wb + 16, 1536);
    v2f b2 = load_b_frag(wb + 32, 1536);
    v2f b3 = load_b_frag(wb + 48, 1536);
    a0 = wmma_f32(a, b0, a0);
    a1 = wmma_f32(a, b1, a1);
    a2 = wmma_f32(a, b2, a2);
    a3 = wmma_f32(a, b3, a3);
  }
  const int n  = lane & 15;
  const int mo = (lane >> 4) * 8;
  v8f acc[4] = {a0, a1, a2, a3};
#pragma unroll
  for (int j = 0; j < 4; ++j) {
    const float bb = bias[n0 + j * 16 + n];
#pragma unroll
    for (int v = 0; v < 8; ++v) {
      float val = acc[j][v] + bb;
      val = 0.5f * val * (1.f + erff(val * 0.70710678118654752f));  // exact gelu
      O[(size_t)(m0 + mo + v) * 1536 + n0 + j * 16 + n] = val;
    }
  }
}

// ---------------- MLP GEMM 2: out = Y + G[16384x1536] @ W[1536x768] + b ----
// Wave tile 16x48 (768 = 16 tiles of 48).
__global__ void mlp_gemm2(const float* __restrict__ A,
                          const float* __restrict__ W,
                          const float* __restrict__ bias,
                          const float* __restrict__ Y,
                          float* __restrict__ O) {
  const int wave = threadIdx.x >> 5;
  const int lane = threadIdx.x & 31;
  const int m0 = blockIdx.x * 16;
  const int n0 = (blockIdx.y * 8 + wave) * 48;
  v8f a0 = {}, a1 = {}, a2 = {};
  const float* arow = A + (size_t)m0 * 1536;
  for (int k = 0; k < 1536; k += 4) {
    __builtin_prefetch(arow + k + 32, 0, 0);
    v2f a = load_a_frag(arow + k, 1536);
    const float* wb = W + (size_t)k * 768 + n0;
    v2f b0 = load_b_frag(wb,      768);
    v2f b1 = load_b_frag(wb + 16, 768);
    v2f b2 = load_b_frag(wb + 32, 768);
    a0 = wmma_f32(a, b0, a0);
    a1 = wmma_f32(a, b1, a1);
    a2 = wmma_f32(a, b2, a2);
  }
  const int n  = lane & 15;
  const int mo = (lane >> 4) * 8;
  v8f acc[3] = {a0, a1, a2};
#pragma unroll
  for (int j = 0; j < 3; ++j) {
    const float bb = bias[n0 + j * 16 + n];
#pragma unroll
    for (int v = 0; v < 8; ++v) {
      const size_t idx = (size_t)(m0 + mo + v) * 768 + n0 + j * 16 + n;
      O[idx] = Y[idx] + acc[j][v] + bb;
    }
  }
}

// ---------------- host-side orchestration ----------------------------------
extern "C" void kernel_launch(void* const* d_in, const int* in_sizes, int n_in,
                              void* d_out, int out_size, void* d_ws, size_t ws_size,
                              hipStream_t stream) {
  const float* x      = (const float*)d_in[0];
  const float* w1     = (const float*)d_in[1];   // (2, 8, 96, 96)
  const float* w2     = (const float*)d_in[2];
  const float* fb1    = (const float*)d_in[3];   // (2, 8, 96)
  const float* fb2    = (const float*)d_in[4];
  const float* ln_g   = (const float*)d_in[5];
  const float* ln_b   = (const float*)d_in[6];
  const float* mlp_w1 = (const float*)d_in[7];   // (768, 1536)
  const float* mlp_b1 = (const float*)d_in[8];
  const float* mlp_w2 = (const float*)d_in[9];   // (1536, 768)
  const float* mlp_b2 = (const float*)d_in[10];
  float* out = (float*)d_out;
  float* ws  = (float*)d_ws;

  const size_t M768 = (size_t)TOKENS * HIDDEN;   // 12.58M floats
  const size_t MF   = (size_t)MROWS  * HIDDEN;   // 6.34M floats
  float* XN  = ws;             // LN output; later reused for second LN
  float* XR  = XN  + M768;     // freq real; later reused for layer-2 real out
  float* XI  = XR  + MF;       // freq imag; later reused for layer-2 imag out
  float* O1R = XI  + MF;
  float* O1I = O1R + MF;
  float* Yb  = O1I + MF;       // residual + filtered signal
  float* Gm  = Yb  + M768;     // MLP hidden (16384 x 1536)

  const dim3 blk(256);
  // 1) xn = LN(x)
  layernorm_kernel<<<dim3(TOKENS / 8), blk, 0, stream>>>(x, ln_g, ln_b, XN);
  // 2) XF = rfft(xn, ortho) per (b, channel)
  fft_fwd_kernel<<<dim3(HIDDEN, BATCH), blk, 0, stream>>>(XN, XR, XI);
  // 3) layer 1: relu(complex block-diag matmul + bias)
  freq_gemm<<<dim3(MROWS / 16, 6), blk, 0, stream>>>(
      XR, XI, w1, w1 + NB * BS * BS, fb1, fb1 + NB * BS, O1R, O1I, 1);
  // 4) layer 2 (no relu), outputs reuse XR/XI
  freq_gemm<<<dim3(MROWS / 16, 6), blk, 0, stream>>>(
      O1R, O1I, w2, w2 + NB * BS * BS, fb2, fb2 + NB * BS, XR, XI, 0);
  // 5) y = x + irfft(YF, ortho)
  fft_inv_kernel<<<dim3(HIDDEN, BATCH), blk, 0, stream>>>(XR, XI, x, Yb);
  // 6) hn = LN(y)  (reuse XN)
  layernorm_kernel<<<dim3(TOKENS / 8), blk, 0, stream>>>(Yb, ln_g, ln_b, XN);
  // 7) G = gelu(hn @ w1 + b1)
  mlp_gemm1<<<dim3(TOKENS / 16, 3), blk, 0, stream>>>(XN, mlp_w1, mlp_b1, Gm);
  // 8) out = y + G @ w2 + b2
  mlp_gemm2<<<dim3(TOKENS / 16, 2), blk, 0, stream>>>(Gm, mlp_w2, mlp_b2, Yb, out);
}